// LocalMamba2D_27453430956043
// MI455X (gfx1250) — compile-verified
//
#include <hip/hip_runtime.h>
#include <hip/hip_bf16.h>
#include <math.h>

// ---------------------------------------------------------------------------
// LocalMamba2D for MI455X (gfx1250, wave32). All f32, working set L2-resident.
//   K1  gn1 stats -> K2 h = groupnorm(x)
//   K3  a = silu(W1 h + b), vpre = W2 h + b     [V_WMMA_F32_16X16X4_F32]
//   K4  v = silu(dwconv3x3(vpre))
//   K5  xd = Wx v (one GEMM shared by all 4 scan directions)   [WMMA]
//   K7  segmented 4-direction selective scan:
//         pass A: per-segment (prod dA, h_out) with h_in = 0
//         fixup : serial chain of 32 segment states per (b,d,c,n)
//         pass B: exact replay from h_in, y accumulated via atomics
//       thread <-> channel; 16 N-states live in registers (no shuffles,
//       delta/softplus computed once per (c,t), B/C are LDS broadcasts)
//   K8/K9 gn2 + gate-mul + channel pool, K10 SE MLP, K11 residual scatter
// ---------------------------------------------------------------------------

typedef float v2f __attribute__((ext_vector_type(2)));
typedef float v8f __attribute__((ext_vector_type(8)));

#define CQ 96
#define LQ 16384     // 128*128 per quadrant
#define NSTATE 16
#define XDROWS 48    // 38 valid rows (R + 2N) padded to 3 WMMA tiles
#define SEG 32       // scan segments per (b, direction)
#define TSEG 512     // timesteps per segment (SEG * TSEG == LQ)
#define TT 64        // LDS staging chunk

__device__ __forceinline__ float silu_f(float v)    { return v / (1.0f + __expf(-v)); }
__device__ __forceinline__ float sigmoid_f(float v) { return 1.0f / (1.0f + __expf(-v)); }

// quadrant (q, c, p) -> flat index into full (1,96,256,256)
__device__ __forceinline__ size_t full_idx(int q, int c, int p) {
  int hh = p >> 7, ww = p & 127;
  int r0 = (q >> 1) * 128, c0 = (q & 1) * 128;
  return ((size_t)c * 256 + (size_t)(r0 + hh)) * 256 + (size_t)(c0 + ww);
}

// ------------------------- K1: groupnorm stats over x -----------------------
__global__ __launch_bounds__(256) void gn_stats_x(const float* __restrict__ x,
                                                  float* __restrict__ stats) {
  const int c = blockIdx.x, q = blockIdx.y, tid = threadIdx.x;
  float s = 0.f, s2 = 0.f;
  for (int p = tid; p < LQ; p += 256) {
    float v = x[full_idx(q, c, p)];
    s += v; s2 += v * v;
  }
  __shared__ float rs_[256], rq_[256];
  rs_[tid] = s; rq_[tid] = s2; __syncthreads();
  for (int off = 128; off > 0; off >>= 1) {
    if (tid < off) { rs_[tid] += rs_[tid + off]; rq_[tid] += rq_[tid + off]; }
    __syncthreads();
  }
  if (tid == 0) {
    atomicAdd(&stats[q * 2 + 0], rs_[0]);
    atomicAdd(&stats[q * 2 + 1], rq_[0]);
  }
}

__global__ void gn_finalize(const float* __restrict__ stats,
                            float* __restrict__ musig, float invCount) {
  int q = threadIdx.x;
  if (q < 4) {
    float mu  = stats[q * 2 + 0] * invCount;
    float var = stats[q * 2 + 1] * invCount - mu * mu;
    musig[q * 2 + 0] = mu;
    musig[q * 2 + 1] = rsqrtf(var + 1e-5f);
  }
}

// ------------------------- K2: apply groupnorm to x -> h --------------------
__global__ __launch_bounds__(256) void gn_apply_x(const float* __restrict__ x,
                                                  const float* __restrict__ musig,
                                                  const float* __restrict__ g,
                                                  const float* __restrict__ bta,
                                                  float* __restrict__ h) {
  const int bc = blockIdx.x, q = bc / CQ, c = bc % CQ;
  const float mu = musig[q * 2], rs = musig[q * 2 + 1];
  const float gg = g[c], bb = bta[c];
  float* dst = h + (size_t)bc * LQ;
  for (int p = threadIdx.x; p < LQ; p += 256)
    dst[p] = (x[full_idx(q, c, p)] - mu) * rs * gg + bb;
}

// ------------------------- K3/K5: GEMM via V_WMMA_F32_16X16X4_F32 -----------
// Y[b,o,l] = act( sum_k Wt[o,k] * X[b,k,l] + bias[o] )
// A (16x4 f32, ISA 7.12.2): lanes 0-15 K=k,k+1 in v0,v1; lanes 16-31 K=k+2,k+3.
// B mirrors: lanes 0-15 rows k,k+1 at N=lane; lanes 16-31 rows k+2,k+3.
// D: VGPR r -> (M=r, N=lane) lanes 0-15; (M=r+8) lanes 16-31.
__global__ __launch_bounds__(256)
void gemm_wmma_f32(const float* __restrict__ X, const float* __restrict__ Wt,
                   const float* __restrict__ bias, float* __restrict__ Y,
                   int K, int Mrows, int Mld, int act) {
  __shared__ float wlds[16][100];
  const int b = blockIdx.z, o0 = blockIdx.y * 16, tid = threadIdx.x;
  for (int e = tid; e < 16 * K; e += 256) {
    int r = e / K, cc = e % K, row = o0 + r;
    wlds[r][cc] = (row < Mrows) ? Wt[(size_t)row * K + cc] : 0.0f;
  }
  __syncthreads();
  const int wave = tid >> 5, lane = tid & 31;
  const int nn = lane & 15, hi = lane >> 4;
  const int l0 = blockIdx.x * 128 + wave * 16;
  const float* xb = X + (size_t)b * K * LQ;
  v8f acc = {};
  for (int k = 0; k < K; k += 4) {
    v2f afrag, bfrag;
    afrag.x = wlds[nn][k + 2 * hi];
    afrag.y = wlds[nn][k + 2 * hi + 1];
    const float* xp = xb + (size_t)(k + 2 * hi) * LQ + l0 + nn;
    bfrag.x = xp[0];
    bfrag.y = xp[LQ];
    acc = __builtin_amdgcn_wmma_f32_16x16x4_f32(false, afrag, false, bfrag,
                                                (short)0, acc, false, false);
  }
  float* yb = Y + (size_t)b * Mld * LQ;
#pragma unroll
  for (int r = 0; r < 8; ++r) {
    int row = o0 + r + 8 * hi;
    float vv = acc[r];
    if (bias && row < Mrows) vv += bias[row];
    if (act) vv = silu_f(vv);
    yb[(size_t)row * LQ + l0 + nn] = vv;
  }
}

// ------------------------- K4: depthwise 3x3 + bias + silu ------------------
__global__ __launch_bounds__(256) void dwconv3x3_silu(const float* __restrict__ vin,
                                                      const float* __restrict__ w,
                                                      const float* __restrict__ bias,
                                                      float* __restrict__ vout) {
  const int bc = blockIdx.x, c = bc % CQ;
  const float* src = vin + (size_t)bc * LQ;
  float* dst = vout + (size_t)bc * LQ;
  float k[9];
#pragma unroll
  for (int i = 0; i < 9; ++i) k[i] = w[c * 9 + i];
  const float bb = bias[c];
  for (int p = threadIdx.x; p < LQ; p += 256) {
    int y = p >> 7, x = p & 127;
    float s = bb;
#pragma unroll
    for (int dy = -1; dy <= 1; ++dy)
#pragma unroll
      for (int dx = -1; dx <= 1; ++dx) {
        int yy = y + dy, xx = x + dx;
        if (yy >= 0 && yy < 128 && xx >= 0 && xx < 128)
          s += k[(dy + 1) * 3 + (dx + 1)] * src[yy * 128 + xx];
      }
    dst[p] = silu_f(s);
  }
}

// ------------------------- K7: segmented selective scan ---------------------
// grid (SEG, dir, b), 96 threads (thread = channel). Per-thread registers hold
// all 16 N-states. pass 0: compute (prod dA, h_out) per segment with h_in = 0.
// pass 1: replay from exact h_in (segIn) and emit y into ysum (atomic across
// the 4 directions; segments within one (b,d) touch disjoint positions).
__global__ __launch_bounds__(96)
void ss2d_scan_pass(const float* __restrict__ v, const float* __restrict__ xd,
                    const float* __restrict__ A_log, const float* __restrict__ dtw,
                    const float* __restrict__ dtb, float* __restrict__ segA,
                    float* __restrict__ segH, const float* __restrict__ segIn,
                    float* __restrict__ ysum, int pass) {
  const int seg = blockIdx.x, d = blockIdx.y, b = blockIdx.z;
  const int c = threadIdx.x;                       // 0..95 = channel
  __shared__ float xs[38][TT + 1];                 // dt rows 0..5, B 6..21, C 22..37
  __shared__ float us[CQ][TT + 1];                 // u per channel (padded: no bank conflicts)
  __shared__ float yt[CQ][TT + 1];

  const float w0 = dtw[c * 6 + 0], w1 = dtw[c * 6 + 1], w2 = dtw[c * 6 + 2];
  const float w3 = dtw[c * 6 + 3], w4 = dtw[c * 6 + 4], w5 = dtw[c * 6 + 5];
  const float bias = dtb[c];

  float acn[NSTATE], hreg[NSTATE], ap[NSTATE];
  const size_t sb = (((size_t)(b * 4 + d) * SEG) + seg) * (CQ * NSTATE) + (size_t)c * NSTATE;
#pragma unroll
  for (int n = 0; n < NSTATE; ++n) {
    acn[n] = -__expf(A_log[c * NSTATE + n]);
    ap[n] = 1.0f;
    hreg[n] = pass ? segIn[sb + n] : 0.0f;
  }

  const float* xdb = xd + (size_t)b * XDROWS * LQ;
  const float* vb  = v  + (size_t)b * CQ * LQ;
  float*       yb  = ysum + (size_t)b * CQ * LQ;

  auto mappos = [&](int t) -> int {
    int tp = (d >= 2) ? (LQ - 1 - t) : t;
    return (d & 1) ? ((tp & 127) * 128 + (tp >> 7)) : tp;
  };
  const int kmax = pass ? 38 : 22;                 // pass A never reads C rows
  auto stage = [&](int gt0) {
    for (int e = c; e < kmax * TT; e += 96) {
      int k = e / TT, t = e % TT;
      xs[k][t] = xdb[(size_t)k * LQ + mappos(gt0 + t)];
    }
    for (int e = c; e < CQ * TT; e += 96) {
      int r = e / TT, t = e % TT;
      us[r][t] = vb[(size_t)r * LQ + mappos(gt0 + t)];
    }
  };

  const int gbase = seg * TSEG;
  stage(gbase);
  __syncthreads();
  for (int cs = 0; cs < TSEG; cs += TT) {
    if (pass == 0) {
      for (int t = 0; t < TT; ++t) {
        float dd = bias + w0 * xs[0][t] + w1 * xs[1][t] + w2 * xs[2][t]
                        + w3 * xs[3][t] + w4 * xs[4][t] + w5 * xs[5][t];
        float delta = (dd > 20.0f) ? dd : log1pf(__expf(dd));
        float du = delta * us[c][t];
#pragma unroll
        for (int n = 0; n < NSTATE; ++n) {
          float dA = __expf(delta * acn[n]);
          ap[n] *= dA;
          hreg[n] = dA * hreg[n] + du * xs[6 + n][t];
        }
      }
    } else {
      for (int t = 0; t < TT; ++t) {
        float dd = bias + w0 * xs[0][t] + w1 * xs[1][t] + w2 * xs[2][t]
                        + w3 * xs[3][t] + w4 * xs[4][t] + w5 * xs[5][t];
        float delta = (dd > 20.0f) ? dd : log1pf(__expf(dd));
        float du = delta * us[c][t];
        float y = 0.0f;
#pragma unroll
        for (int n = 0; n < NSTATE; ++n) {
          float dA = __expf(delta * acn[n]);
          hreg[n] = dA * hreg[n] + du * xs[6 + n][t];
          y += hreg[n] * xs[22 + n][t];
        }
        yt[c][t] = y;
      }
    }
    __syncthreads();
    if (pass) {
      for (int e = c; e < CQ * TT; e += 96) {
        int r = e >> 6, t = e & 63;                // consecutive lanes -> consecutive t
        atomicAdd(&yb[(size_t)r * LQ + mappos(gbase + cs + t)], yt[r][t]);
      }
    }
    if (cs + TT < TSEG) stage(gbase + cs + TT);
    __syncthreads();
  }
  if (pass == 0) {
#pragma unroll
    for (int n = 0; n < NSTATE; ++n) { segA[sb + n] = ap[n]; segH[sb + n] = hreg[n]; }
  }
}

// serial chain of segment states: h_in[s] = A[s-1]*h_in[s-1] + H[s-1]
__global__ __launch_bounds__(256)
void scan_fixup(const float* __restrict__ segA, const float* __restrict__ segH,
                float* __restrict__ segIn) {
  const int idx = blockIdx.x * 256 + threadIdx.x;  // (b*4+d)*1536 + c*16+n
  if (idx >= 16 * CQ * NSTATE) return;
  const int bd = idx / (CQ * NSTATE), cn = idx % (CQ * NSTATE);
  float h = 0.0f;
  for (int s = 0; s < SEG; ++s) {
    size_t off = ((size_t)bd * SEG + s) * (CQ * NSTATE) + cn;
    segIn[off] = h;
    h = segA[off] * h + segH[off];
  }
}

// ------------------------- K8: stats over (ysum + 4*Dp*v) -------------------
__global__ __launch_bounds__(256) void gn_stats_y(const float* __restrict__ ysum,
                                                  const float* __restrict__ v,
                                                  const float* __restrict__ Dp,
                                                  float* __restrict__ stats) {
  const int c = blockIdx.x, q = blockIdx.y, tid = threadIdx.x;
  const size_t base = ((size_t)q * CQ + c) * LQ;
  const float dp4 = 4.0f * Dp[c];
  float s = 0.f, s2 = 0.f;
  for (int p = tid; p < LQ; p += 256) {
    float val = ysum[base + p] + dp4 * v[base + p];
    s += val; s2 += val * val;
  }
  __shared__ float rs_[256], rq_[256];
  rs_[tid] = s; rq_[tid] = s2; __syncthreads();
  for (int off = 128; off > 0; off >>= 1) {
    if (tid < off) { rs_[tid] += rs_[tid + off]; rq_[tid] += rq_[tid + off]; }
    __syncthreads();
  }
  if (tid == 0) {
    atomicAdd(&stats[q * 2 + 0], rs_[0]);
    atomicAdd(&stats[q * 2 + 1], rq_[0]);
  }
}

// ------------------------- K9: normalize, gate-mul, channel pool ------------
__global__ __launch_bounds__(256)
void norm_mul_pool(const float* __restrict__ ysum, const float* __restrict__ v,
                   const float* __restrict__ a, const float* __restrict__ musig,
                   const float* __restrict__ Dp, const float* __restrict__ n2g,
                   const float* __restrict__ n2b, float* __restrict__ y2,
                   float* __restrict__ spool) {
  const int bc = blockIdx.x, q = bc / CQ, c = bc % CQ, tid = threadIdx.x;
  const size_t base = (size_t)bc * LQ;
  const float mu = musig[q * 2], rs = musig[q * 2 + 1];
  const float gg = n2g[c], bb = n2b[c], dp4 = 4.0f * Dp[c];
  float local = 0.f;
  for (int p = tid; p < LQ; p += 256) {
    float bf = ysum[base + p] + dp4 * v[base + p];
    float bn = (bf - mu) * rs * gg + bb;
    float yv = a[base + p] * bn;
    y2[base + p] = yv;
    local += yv;
  }
  __shared__ float rs_[256];
  rs_[tid] = local; __syncthreads();
  for (int off = 128; off > 0; off >>= 1) {
    if (tid < off) rs_[tid] += rs_[tid + off];
    __syncthreads();
  }
  if (tid == 0) spool[bc] = rs_[0] * (1.0f / (float)LQ);
}

// ------------------------- K10: SE gate (96 -> 12 -> 96) --------------------
__global__ void se_gate(const float* __restrict__ s, const float* __restrict__ w1,
                        const float* __restrict__ b1, const float* __restrict__ w2,
                        const float* __restrict__ b2, float* __restrict__ gate) {
  const int q = blockIdx.x, t = threadIdx.x;
  __shared__ float sv[96], hid[12];
  if (t < 96) sv[t] = s[q * 96 + t];
  __syncthreads();
  if (t < 12) {
    float acc = b1[t];
    for (int c2 = 0; c2 < 96; ++c2) acc += sv[c2] * w1[t * 96 + c2];
    hid[t] = fmaxf(acc, 0.0f);
  }
  __syncthreads();
  if (t < 96) {
    float acc = b2[t];
    for (int j = 0; j < 12; ++j) acc += hid[j] * w2[t * 12 + j];
    gate[q * 96 + t] = sigmoid_f(acc);
  }
}

// ------------------------- K11: residual + scatter to quadrants -------------
__global__ __launch_bounds__(256)
void final_out(const float* __restrict__ x, const float* __restrict__ y2,
               const float* __restrict__ gate, float* __restrict__ out) {
  const int bc = blockIdx.x, q = bc / CQ, c = bc % CQ;
  const float g = gate[bc];
  const float* src = y2 + (size_t)bc * LQ;
  for (int p = threadIdx.x; p < LQ; p += 256) {
    size_t gi = full_idx(q, c, p);
    out[gi] = x[gi] + src[p] * g;
  }
}

// ---------------------------------------------------------------------------
extern "C" void kernel_launch(void* const* d_in, const int* in_sizes, int n_in,
                              void* d_out, int out_size, void* d_ws, size_t ws_size,
                              hipStream_t stream) {
  const float* x        = (const float*)d_in[0];
  const float* ln_g     = (const float*)d_in[1];
  const float* ln_b     = (const float*)d_in[2];
  const float* p1_w     = (const float*)d_in[3];
  const float* p1_b     = (const float*)d_in[4];
  const float* p2_in_w  = (const float*)d_in[5];
  const float* p2_in_b  = (const float*)d_in[6];
  const float* p2_dw_w  = (const float*)d_in[7];
  const float* p2_dw_b  = (const float*)d_in[8];
  const float* A_log    = (const float*)d_in[9];
  const float* Dp       = (const float*)d_in[10];
  const float* x_proj_w = (const float*)d_in[11];
  const float* dt_proj_w= (const float*)d_in[12];
  const float* dt_proj_b= (const float*)d_in[13];
  const float* n2_g     = (const float*)d_in[14];
  const float* n2_b     = (const float*)d_in[15];
  const float* ca_w1    = (const float*)d_in[16];
  const float* ca_b1    = (const float*)d_in[17];
  const float* ca_w2    = (const float*)d_in[18];
  const float* ca_b2    = (const float*)d_in[19];
  float* out = (float*)d_out;
  float* W   = (float*)d_ws;

  // workspace layout (floats)
  const size_t PLANE  = (size_t)CQ * LQ;           // per-quadrant channel plane set
  const size_t SEGTOT = (size_t)16 * SEG * CQ * NSTATE; // 786,432
  float* stats1 = W + 0;                           // 8
  float* musig1 = W + 8;                           // 8
  float* stats2 = W + 16;                          // 8
  float* musig2 = W + 24;                          // 8
  float* spool  = W + 32;                          // 384
  float* gatebf = W + 512;                         // 384
  float* h      = W + 1024;                        // 4*PLANE
  float* abuf   = h    + 4 * PLANE;                // 4*PLANE
  float* vpre   = abuf + 4 * PLANE;                // 4*PLANE
  float* vbuf   = vpre + 4 * PLANE;                // 4*PLANE
  float* xd     = vbuf + 4 * PLANE;                // 4*XDROWS*LQ
  float* segA   = xd   + (size_t)4 * XDROWS * LQ;  // SEGTOT
  float* segH   = segA + SEGTOT;                   // SEGTOT
  float* segIn  = segH + SEGTOT;                   // SEGTOT
  float* ysum   = h;                               // reuse: h dead after GEMMs
  float* y2     = vpre;                            // reuse: vpre dead after dwconv

  const float invCL = 1.0f / (float)(CQ * LQ);

  // zero atomic accumulators (graph-replay deterministic)
  hipMemsetAsync(W, 0, 1024 * sizeof(float), stream);

  gn_stats_x<<<dim3(CQ, 4), 256, 0, stream>>>(x, stats1);
  gn_finalize<<<1, 4, 0, stream>>>(stats1, musig1, invCL);
  gn_apply_x<<<4 * CQ, 256, 0, stream>>>(x, musig1, ln_g, ln_b, h);

  // a = silu(p1 @ h + b)   [WMMA]
  gemm_wmma_f32<<<dim3(128, 6, 4), 256, 0, stream>>>(h, p1_w, p1_b, abuf,
                                                     CQ, CQ, CQ, /*act=*/1);
  // vpre = p2_in @ h + b   [WMMA]
  gemm_wmma_f32<<<dim3(128, 6, 4), 256, 0, stream>>>(h, p2_in_w, p2_in_b, vpre,
                                                     CQ, CQ, CQ, /*act=*/0);
  // v = silu(dwconv3x3(vpre))
  dwconv3x3_silu<<<4 * CQ, 256, 0, stream>>>(vpre, p2_dw_w, p2_dw_b, vbuf);
  // xd = x_proj @ v (shared across all 4 scan directions; pointwise in l) [WMMA]
  gemm_wmma_f32<<<dim3(128, 3, 4), 256, 0, stream>>>(vbuf, x_proj_w, nullptr, xd,
                                                     CQ, 38, XDROWS, /*act=*/0);

  hipMemsetAsync(ysum, 0, 4 * PLANE * sizeof(float), stream);
  // segmented exact scan: pass A -> fixup -> pass B
  ss2d_scan_pass<<<dim3(SEG, 4, 4), 96, 0, stream>>>(vbuf, xd, A_log, dt_proj_w,
                                                     dt_proj_b, segA, segH, segIn,
                                                     ysum, /*pass=*/0);
  scan_fixup<<<96, 256, 0, stream>>>(segA, segH, segIn);
  ss2d_scan_pass<<<dim3(SEG, 4, 4), 96, 0, stream>>>(vbuf, xd, A_log, dt_proj_w,
                                                     dt_proj_b, segA, segH, segIn,
                                                     ysum, /*pass=*/1);

  gn_stats_y<<<dim3(CQ, 4), 256, 0, stream>>>(ysum, vbuf, Dp, stats2);
  gn_finalize<<<1, 4, 0, stream>>>(stats2, musig2, invCL);
  norm_mul_pool<<<4 * CQ, 256, 0, stream>>>(ysum, vbuf, abuf, musig2, Dp,
                                            n2_g, n2_b, y2, spool);
  se_gate<<<4, 96, 0, stream>>>(spool, ca_w1, ca_b1, ca_w2, ca_b2, gatebf);
  final_out<<<4 * CQ, 256, 0, stream>>>(x, y2, gatebf, out);
}